// DDSP_30296699306258
// MI455X (gfx1250) — compile-verified
//
#include <hip/hip_runtime.h>
#include <math.h>

typedef __attribute__((ext_vector_type(2))) float v2f;
typedef __attribute__((ext_vector_type(8))) float v8f;

#define NBATCH 4
#define NT 1000
#define H1 101
#define NH 100
#define NBANDS 65
#define BLK 256
#define SRATE 16000
#define NS (NT*BLK)              // 256000 samples per batch
#define XPAD 16
#define XSTRIDE (NS + 2*XPAD)    // 256032
#define HLEN 16000
#define HBUF_LEN 16048
#define HR_BASE 16014            // hr[HR_BASE - j] = h[j];  j in (-34, 16014]

// workspace layout (float offsets)
#define AMPS_OFF 0                           // 4000*100
#define DPHI_OFF (AMPS_OFF + NBATCH*NT*NH)   // 400000
#define BASE_OFF (DPHI_OFF + NBATCH*NT)      // 404000
#define HBUF_OFF (BASE_OFF + NBATCH*NT)      // 408000
#define XBUF_OFF (HBUF_OFF + HBUF_LEN)       // 424048  (+ 4*256032)

__device__ __forceinline__ float scale_fn(float x) {
    float s = 1.0f / (1.0f + expf(-x));
    return 2.0f * powf(s, 2.302585092994046f) + 1e-7f;  // 2*sigmoid^ln(10)+1e-7
}

// ---------------- K1: per-frame amplitude shaping ----------------
__global__ void k1_frame(const float* __restrict__ pitch,
                         const float* __restrict__ amp_param,
                         float* __restrict__ ws) {
    __shared__ float red[128];
    __shared__ float s_total;
    int fr = blockIdx.x;              // b*1000 + f
    int tid = threadIdx.x;
    float p = pitch[fr];
    float val = 0.0f;
    if (tid < H1) val = scale_fn(amp_param[fr * H1 + tid]);
    if (tid == 0) s_total = val;
    float masked = 0.0f;
    if (tid >= 1 && tid < H1) {
        float aa = ((p * (float)tid) < 8000.0f ? 1.0f : 0.0f) + 1e-4f;
        masked = val * aa;
    }
    red[tid] = masked;
    __syncthreads();
    for (int s = 64; s > 0; s >>= 1) {
        if (tid < s) red[tid] += red[tid + s];
        __syncthreads();
    }
    float sum = red[0];
    if (tid >= 1 && tid < H1)
        ws[AMPS_OFF + fr * NH + (tid - 1)] = masked / sum * s_total;
    if (tid == 0)
        ws[DPHI_OFF + fr] = 6.2831853071795864f * p / (float)SRATE;
}

// ---------------- K2: per-batch phase prefix over frames ----------------
__global__ void k2_base(float* __restrict__ ws) {
    int b = threadIdx.x;
    if (b < NBATCH) {
        float run = 0.0f;
        for (int f = 0; f < NT; ++f) {
            ws[BASE_OFF + b * NT + f] = run;
            run += 256.0f * ws[DPHI_OFF + b * NT + f];
        }
    }
}

// ---------------- K4: reversed reverb impulse + signal guard zeroing ----------------
__global__ void k4_prep(const float* __restrict__ rev_noise,
                        const float* __restrict__ rev_decay,
                        const float* __restrict__ rev_wet,
                        float* __restrict__ ws) {
    int i = blockIdx.x * blockDim.x + threadIdx.x;
    float* hr = ws + HBUF_OFF;
    float* xb = ws + XBUF_OFF;
    if (i < HBUF_LEN) {
        int j = HR_BASE - i;          // tap index stored (reversed layout)
        float v = 0.0f;
        if (j >= 0 && j < HLEN) {
            if (j == 0) v = 1.0f;
            else {
                float sp  = log1pf(expf(-rev_decay[0]));       // softplus(-decay)
                float tt  = (float)j / (float)SRATE;
                float env = expf(-sp * tt * 500.0f);
                float wet = 1.0f / (1.0f + expf(-rev_wet[0])); // sigmoid
                v = rev_noise[j] * env * wet;
            }
        }
        hr[i] = v;
    } else if (i < HBUF_LEN + NBATCH * 2 * XPAD) {
        int g = i - HBUF_LEN;      // 0..127
        int bb = g >> 5;
        int r  = g & 31;
        int pos = (r < XPAD) ? r : (XPAD + NS + (r - XPAD));
        xb[bb * XSTRIDE + pos] = 0.0f;
    }
}

// ---------------- K3: harmonic synth + filtered-noise per frame ----------------
__global__ void k3_signal(const float* __restrict__ noise_param,
                          const float* __restrict__ noise_u,
                          float* __restrict__ ws) {
    __shared__ float s_np[NBANDS];
    __shared__ float s_ir[128];
    __shared__ float s_kern[256];
    __shared__ float s_nz[256];
    __shared__ float s_amps[NH];
    int fr = blockIdx.x;
    int tid = threadIdx.x;
    if (tid < NBANDS) s_np[tid] = scale_fn(noise_param[fr * NBANDS + tid] - 5.0f);
    if (tid < NH)     s_amps[tid] = ws[AMPS_OFF + fr * NH + tid];
    s_nz[tid] = noise_u[fr * 256 + tid] * 2.0f - 1.0f;
    __syncthreads();
    if (tid < 128) {   // irfft of real 65-bin spectrum -> 128 taps
        float c1 = cospif((float)tid * (1.0f / 64.0f));
        float acc = s_np[0];
        float cp = 1.0f, cc = c1;
        for (int k = 1; k < 64; ++k) {
            acc += 2.0f * s_np[k] * cc;
            float cn = 2.0f * c1 * cc - cp;
            cp = cc; cc = cn;
        }
        acc += s_np[64] * ((tid & 1) ? -1.0f : 1.0f);   // cos(pi*n)
        s_ir[tid] = acc * (1.0f / 128.0f);
    }
    __syncthreads();
    {   // roll + Hann window + pad + roll => sparse 256-tap kernel
        int t = tid;
        float kv = 0.0f;
        if (t < 64)
            kv = 0.5f * (1.0f - cospif((float)(t + 64) * (1.0f / 64.0f))) * s_ir[t];
        else if (t >= 192)
            kv = 0.5f * (1.0f - cospif((float)(t - 192) * (1.0f / 64.0f))) * s_ir[t - 128];
        s_kern[t] = kv;
    }
    __syncthreads();
    int t = tid;
    float ns = 0.0f;
    {   // causal conv; kernel non-zero only on [0,64) U [192,256)
        int jmax = t < 63 ? t : 63;
        for (int j = 0; j <= jmax; ++j) ns += s_kern[j] * s_nz[t - j];
        if (t >= 192)
            for (int j = 192; j <= t; ++j) ns += s_kern[j] * s_nz[t - j];
    }
    float base  = ws[BASE_OFF + fr];
    float dphi  = ws[DPHI_OFF + fr];
    float omega = base + (float)(t + 1) * dphi;
    float s1, c1;
    sincosf(omega, &s1, &c1);
    float hs = 0.0f;
    float sp = 0.0f, sc = s1;
    float twoc = 2.0f * c1;
    for (int h = 1; h <= NH; ++h) {
        hs += sc * s_amps[h - 1];
        float sn = twoc * sc - sp;
        sp = sc; sc = sn;
    }
    int bb = fr / NT, f = fr % NT;
    ws[XBUF_OFF + bb * XSTRIDE + XPAD + f * BLK + t] = hs + ns;
}

// ---------------- K5: 16000-tap reverb FIR as fp32 WMMA GEMM ----------------
// Mapping (reversed-K): A[m][k] = h[j0+m+3-k]   (reversed filter => ascending loads)
//                       B[k][n] = x_n[tb-j0-3+k]           (ascending loads)
// D[m][n] accumulates out_n[tb+m]; j0 walks the step-4 tap grid from -16.

struct OpSet { v2f a0, a1, a2, a3, b0, b1, b2, b3; };

__device__ __forceinline__ void load_set(const float* __restrict__ pa,
                                         const float* __restrict__ pb,
                                         int j0, OpSet& S) {
    const float* A  = pa - j0;     // one base, immediate offsets
    const float* Bp = pb - j0;
    S.a0.x = A[0];    S.a0.y = A[1];    S.b0.x = Bp[0];    S.b0.y = Bp[1];
    S.a1.x = A[-4];   S.a1.y = A[-3];   S.b1.x = Bp[-4];   S.b1.y = Bp[-3];
    S.a2.x = A[-8];   S.a2.y = A[-7];   S.b2.x = Bp[-8];   S.b2.y = Bp[-7];
    S.a3.x = A[-12];  S.a3.y = A[-11];  S.b3.x = Bp[-12];  S.b3.y = Bp[-11];
}

__device__ __forceinline__ void wmma4(const OpSet& S, v8f& c0, v8f& c1, v8f& c2, v8f& c3) {
    c0 = __builtin_amdgcn_wmma_f32_16x16x4_f32(false, S.a0, false, S.b0, (short)0, c0, false, false);
    c1 = __builtin_amdgcn_wmma_f32_16x16x4_f32(false, S.a1, false, S.b1, (short)0, c1, false, false);
    c2 = __builtin_amdgcn_wmma_f32_16x16x4_f32(false, S.a2, false, S.b2, (short)0, c2, false, false);
    c3 = __builtin_amdgcn_wmma_f32_16x16x4_f32(false, S.a3, false, S.b3, (short)0, c3, false, false);
}

// Guarded step, valid for any j0 on the grid (edge clamping via select)
__device__ __forceinline__ void guard_step(const float* __restrict__ hr,
                                           const float* __restrict__ xb,
                                           int tb, int j0, int m, int kk, v8f& acc) {
    v2f a, b;
    int ai = HR_BASE - 3 - j0 - m + kk;   // always within [0, 16031)
    a.x = hr[ai];
    a.y = hr[ai + 1];
    int t1 = tb - j0 - 3 + kk;
    unsigned u1 = (unsigned)(t1 + XPAD);
    unsigned u2 = (unsigned)(t1 + 1 + XPAD);
    float v1 = xb[u1 < (unsigned)XSTRIDE ? u1 : 0u];
    float v2 = xb[u2 < (unsigned)XSTRIDE ? u2 : 0u];
    b.x = (u1 < (unsigned)XSTRIDE) ? v1 : 0.0f;
    b.y = (u2 < (unsigned)XSTRIDE) ? v2 : 0.0f;
    acc = __builtin_amdgcn_wmma_f32_16x16x4_f32(false, a, false, b,
                                                (short)0, acc, false, false);
}

__global__ void k5_reverb(const float* __restrict__ ws, float* __restrict__ out) {
    const float* hr    = ws + HBUF_OFF;
    const float* xbase = ws + XBUF_OFF;
    int wave  = threadIdx.x >> 5;
    int lane  = threadIdx.x & 31;
    int chain = blockIdx.x * (blockDim.x >> 5) + wave;   // 4000 chains
    int t0 = chain * 64;
    // A (16x4 f32): lane = M, VGPR pair = K {0,1} (lanes 0-15) / {2,3} (lanes 16-31)
    int m  = lane & 15;
    int kk = (lane >> 4) << 1;
    // B (4x16 f32): lane%16 = N; N packs (batch, time-tile)
    int n    = lane & 15;
    int bb   = n & 3;
    int tile = n >> 2;
    int tb = t0 + tile * 16;
    const float* xb = xbase + bb * XSTRIDE;
    const float* pa = hr + (HR_BASE - 3) - m + kk;   // A pair at pa[-j0], pa[-j0+1]
    const float* pb = xb + XPAD + tb - 3 + kk;       // B pair at pb[-j0], pb[-j0+1]

    v8f acc0 = {}, acc1 = {}, acc2 = {}, acc3 = {};
    int full_end = t0 + 48;                  // last j0 (inclusive) on step-4 grid
    if (full_end > HLEN - 4) full_end = HLEN - 4;
    int safe_end = t0 + 12;                  // B index >= -XPAD guaranteed
    if (safe_end > full_end) safe_end = full_end;   // >= 12 always

    OpSet X, Y;
    int j0 = -16;
    load_set(pa, pb, j0, X);                 // always in-bounds
    for (; j0 + 28 <= safe_end; j0 += 32) {  // double-buffered, 8 WMMAs / iter
        load_set(pa, pb, j0 + 16, Y);
        __builtin_amdgcn_sched_barrier(0);
        wmma4(X, acc0, acc1, acc2, acc3);
        int jx = j0 + 32;
        if (jx > safe_end - 12) jx = safe_end - 12;  // clamped prefetch (>= 0)
        load_set(pa, pb, jx, X);
        __builtin_amdgcn_sched_barrier(0);
        wmma4(Y, acc0, acc1, acc2, acc3);
    }
    for (; j0 <= full_end; j0 += 4)          // guarded edge iterations
        guard_step(hr, xb, tb, j0, m, kk, acc0);

    v8f acc = (acc0 + acc1) + (acc2 + acc3);
    // D (16x16 f32): VGPR r -> M = r + (lane>=16 ? 8 : 0), N = lane%16
    int mrow = (lane >> 4) << 3;
    float* op = out + bb * NS + tb + mrow;
    *(float4*)(op)     = make_float4(acc[0], acc[1], acc[2], acc[3]);
    *(float4*)(op + 4) = make_float4(acc[4], acc[5], acc[6], acc[7]);
}

extern "C" void kernel_launch(void* const* d_in, const int* in_sizes, int n_in,
                              void* d_out, int out_size, void* d_ws, size_t ws_size,
                              hipStream_t stream) {
    const float* pitch       = (const float*)d_in[0];
    const float* amp_param   = (const float*)d_in[1];
    const float* noise_param = (const float*)d_in[2];
    const float* noise_u     = (const float*)d_in[3];
    const float* rev_noise   = (const float*)d_in[4];
    const float* rev_decay   = (const float*)d_in[5];
    const float* rev_wet     = (const float*)d_in[6];
    float* ws  = (float*)d_ws;
    float* out = (float*)d_out;

    k1_frame <<<NBATCH * NT, 128, 0, stream>>>(pitch, amp_param, ws);
    k2_base  <<<1, 32, 0, stream>>>(ws);
    k4_prep  <<<(HBUF_LEN + NBATCH * 2 * XPAD + 255) / 256, 256, 0, stream>>>(
                 rev_noise, rev_decay, rev_wet, ws);
    k3_signal<<<NBATCH * NT, 256, 0, stream>>>(noise_param, noise_u, ws);
    k5_reverb<<<500, 256, 0, stream>>>(ws, out);   // 500 blocks * 8 waves = 4000 chains
}